// LocSE_10273561772200
// MI455X (gfx1250) — compile-verified
//
#include <hip/hip_runtime.h>
#include <hip/hip_bf16.h>

typedef __attribute__((ext_vector_type(2))) float v2f;
typedef __attribute__((ext_vector_type(8))) float v8f;

#define BB 2
#define NN 65536
#define KK 16
#define DOUT 64
#define DFEAT 64

// ---------------- Phase 1: sp moments (S1[10], S2 upper-tri 55) ----------------
__global__ __launch_bounds__(256) void locse_moments(
    const float* __restrict__ coords, const int* __restrict__ kp,
    const float* __restrict__ kd, double* __restrict__ S) {
  __shared__ float ls[65];
  int t = threadIdx.x;
  if (t < 65) ls[t] = 0.0f;
  __syncthreads();

  int pid = blockIdx.x * 256 + t;              // 0 .. B*N-1 (grid sized exactly)
  int b = pid >> 16;                           // N = 65536
  float acc1[10];
  float acc2[55];
#pragma unroll
  for (int j = 0; j < 10; ++j) acc1[j] = 0.0f;
#pragma unroll
  for (int q = 0; q < 55; ++q) acc2[q] = 0.0f;

  const float* cp = coords + (size_t)pid * 3;
  float cx = cp[0], cy = cp[1], cz = cp[2];
#pragma unroll 4
  for (int k = 0; k < KK; ++k) {
    int idx = kp[(size_t)pid * KK + k];
    const float* np = coords + ((size_t)b * NN + idx) * 3;
    float nx = np[0], ny = np[1], nz = np[2];
    float d = kd[(size_t)pid * KK + k];
    float sp[10] = {cx, cy, cz, nx, ny, nz, cx - nx, cy - ny, cz - nz, d};
    int q = 0;
#pragma unroll
    for (int j = 0; j < 10; ++j) {
      acc1[j] += sp[j];
#pragma unroll
      for (int l = j; l < 10; ++l) { acc2[q] += sp[j] * sp[l]; ++q; }
    }
  }
#pragma unroll
  for (int j = 0; j < 10; ++j) atomicAdd(&ls[j], acc1[j]);        // ds_add_f32
#pragma unroll
  for (int q = 0; q < 55; ++q) atomicAdd(&ls[10 + q], acc2[q]);
  __syncthreads();
  if (t < 65) atomicAdd(&S[t], (double)ls[t]);                    // global_atomic_add_f64
}

// ---------------- Phase 2: fold BatchNorm into W', b' ----------------
__global__ void locse_stats(const double* __restrict__ S,
                            const float* __restrict__ W, const float* __restrict__ bias,
                            const float* __restrict__ gamma, const float* __restrict__ beta,
                            float* __restrict__ Wp, float* __restrict__ bp) {
  int o = threadIdx.x;  // 64 threads
  const double M = (double)BB * NN * KK;
  double mu[10], wr[10];
#pragma unroll
  for (int j = 0; j < 10; ++j) { mu[j] = S[j] / M; wr[j] = (double)W[o * 10 + j]; }
  double bo = (double)bias[o];
  double mean = bo;
#pragma unroll
  for (int j = 0; j < 10; ++j) mean += wr[j] * mu[j];
  double ex2 = 0.0;
  int q = 0;
#pragma unroll
  for (int j = 0; j < 10; ++j)
#pragma unroll
    for (int l = j; l < 10; ++l) {
      double e2 = S[10 + q] / M;
      ex2 += (j == l ? 1.0 : 2.0) * wr[j] * wr[l] * e2;
      ++q;
    }
  ex2 += 2.0 * bo * (mean - bo) + bo * bo;          // E[(W.sp+b)^2]
  double var = ex2 - mean * mean;
  double scale = (double)gamma[o] / sqrt(var + 1e-6);
#pragma unroll
  for (int j = 0; j < 10; ++j) Wp[o * 12 + j] = (float)(scale * wr[j]);
  Wp[o * 12 + 10] = 0.0f;
  Wp[o * 12 + 11] = 0.0f;
  bp[o] = (float)(scale * (bo - mean) + (double)beta[o]);
}

// ---------------- Phase 3: main pass, V_WMMA_F32_16X16X4_F32 ----------------
// Per wave: one point n. out[64,16] = W'[64,12] @ sp[12,16] (+b' via C-init), ReLU, store.
__global__ __launch_bounds__(256) void locse_wmma(
    const float* __restrict__ coords, const int* __restrict__ kp,
    const float* __restrict__ kd, const float* __restrict__ Wp,
    const float* __restrict__ bp, float* __restrict__ out) {
  int lane = threadIdx.x & 31;
  int pid = blockIdx.x * 8 + (threadIdx.x >> 5);   // global wave id == point id
  int b = pid >> 16;
  int n = pid & (NN - 1);
  int k = lane & 15;      // neighbor index (B/C column)
  int h = lane >> 4;      // half-wave select
  int m = lane & 15;      // A-matrix row

  // A fragments: 16x4 f32 layout -> VGPR0 holds K=2h, VGPR1 holds K=2h+1 per lane
  v2f A[4][3];
#pragma unroll
  for (int t = 0; t < 4; ++t)
#pragma unroll
    for (int s = 0; s < 3; ++s) {
      const float* wrow = Wp + (t * 16 + m) * 12 + s * 4 + 2 * h;
      A[t][s].x = wrow[0];
      A[t][s].y = wrow[1];
    }

  // C init with fused bias: VGPR v, lane group h -> channel t*16 + v + 8h
  v8f c[4];
#pragma unroll
  for (int t = 0; t < 4; ++t)
#pragma unroll
    for (int v = 0; v < 8; ++v) c[t][v] = bp[t * 16 + v + 8 * h];

  // Gather spatial encoding for this point / neighbor k
  const float* cp = coords + (size_t)pid * 3;
  float cx = cp[0], cy = cp[1], cz = cp[2];
  int idx = kp[(size_t)pid * KK + k];
  const float* np = coords + ((size_t)b * NN + idx) * 3;
  float nx = np[0], ny = np[1], nz = np[2];
  float d = kd[(size_t)pid * KK + k];
  float rx = cx - nx, ry = cy - ny, rz = cz - nz;

  // B fragments (4x16 f32): VGPR0 = row 2h, VGPR1 = row 2h+1 of the K-step slab
  v2f Bf[3];
  Bf[0].x = h ? cz : cx;  Bf[0].y = h ? nx : cy;    // rows 0..3: cx,cy,cz,nx
  Bf[1].x = h ? rx : ny;  Bf[1].y = h ? ry : nz;    // rows 4..7: ny,nz,rx,ry
  Bf[2].x = h ? 0.f : rz; Bf[2].y = h ? 0.f : d;    // rows 8..11: rz,d,0,0

#pragma unroll
  for (int t = 0; t < 4; ++t)
#pragma unroll
    for (int s = 0; s < 3; ++s)
      c[t] = __builtin_amdgcn_wmma_f32_16x16x4_f32(
          false, A[t][s], false, Bf[s], (short)0, c[t], false, false);

  // ReLU + store: channel ch = t*16 + v + 8h, column = k
#pragma unroll
  for (int t = 0; t < 4; ++t)
#pragma unroll
    for (int v = 0; v < 8; ++v) {
      int ch = t * 16 + v + 8 * h;
      float val = fmaxf(c[t][v], 0.0f);
      out[(((size_t)(b * 128 + ch)) * NN + n) * KK + k] = val;
    }
}

// ---------------- Phase 4: feature broadcast (channels 64..127), float4 stores ----
__global__ __launch_bounds__(256) void locse_feat(const float* __restrict__ feats,
                                                  float* __restrict__ out) {
  size_t t = (size_t)blockIdx.x * 256 + threadIdx.x;  // 0 .. B*64*N-1
  int n = (int)(t & (NN - 1));
  int bf = (int)(t >> 16);          // b*64 + f
  int b = bf >> 6;
  int f = bf & 63;
  float val = feats[t];             // features flat layout [B,64,N,1] == t
  float4 v4 = make_float4(val, val, val, val);
  float4* o = (float4*)(out + (((size_t)(b * 128 + 64 + f)) * NN + n) * KK);
  o[0] = v4; o[1] = v4; o[2] = v4; o[3] = v4;
}

extern "C" void kernel_launch(void* const* d_in, const int* in_sizes, int n_in,
                              void* d_out, int out_size, void* d_ws, size_t ws_size,
                              hipStream_t stream) {
  const float* coords = (const float*)d_in[0];
  const float* feats  = (const float*)d_in[1];
  const int*   kp     = (const int*)d_in[2];
  const float* kd     = (const float*)d_in[3];
  const float* W      = (const float*)d_in[4];
  const float* bias   = (const float*)d_in[5];
  const float* gamma  = (const float*)d_in[6];
  const float* beta   = (const float*)d_in[7];
  float* out = (float*)d_out;

  double* S  = (double*)d_ws;                         // 65 doubles
  float*  Wp = (float*)((char*)d_ws + 1024);          // 64*12 floats
  float*  bp = (float*)((char*)d_ws + 1024 + 64 * 12 * sizeof(float));

  hipMemsetAsync(d_ws, 0, 65 * sizeof(double), stream);

  locse_moments<<<(BB * NN) / 256, 256, 0, stream>>>(coords, kp, kd, S);
  locse_stats<<<1, 64, 0, stream>>>(S, W, bias, gamma, beta, Wp, bp);
  locse_wmma<<<(BB * NN) / 8, 256, 0, stream>>>(coords, kp, kd, Wp, bp, out);
  locse_feat<<<(BB * DFEAT * NN) / 256, 256, 0, stream>>>(feats, out);
}